// GNNEmbedder_24678882083279
// MI455X (gfx1250) — compile-verified
//
#include <hip/hip_runtime.h>
#include <hip/hip_bf16.h>

#define N_NODES   50000
#define N_EDGES   800000
#define N_GRAPHS  128
#define DHID      64
#define NEG_SLOPE 0.2f

typedef float v2f __attribute__((ext_vector_type(2)));
typedef float v8f __attribute__((ext_vector_type(8)));

// ---------------------------------------------------------------------------
// GEMM: H[N,64] = X[N,64] @ W[64,64] using V_WMMA_F32_16X16X4_F32.
// One wave computes a 16-row x 64-col slab; 16 K-steps x 4 column tiles.
// ---------------------------------------------------------------------------
__global__ __launch_bounds__(256)
void gat_gemm_wmma(const float* __restrict__ X, const float* __restrict__ W,
                   float* __restrict__ H, int n)
{
    const int waveInBlk = threadIdx.x >> 5;
    const int lane      = threadIdx.x & 31;
    const int waveGlob  = blockIdx.x * 8 + waveInBlk;
    const int row0      = waveGlob * 16;
    const int lrow      = lane & 15;
    const int hi        = lane >> 4;          // 0: K pair {0,1}, 1: K pair {2,3}

    int rowA = row0 + lrow;
    if (rowA > n - 1) rowA = n - 1;           // clamp: keep EXEC all-ones for WMMA
    const float* __restrict__ xr = X + (size_t)rowA * DHID;

    v8f acc0 = {}, acc1 = {}, acc2 = {}, acc3 = {};

#pragma unroll
    for (int k0 = 0; k0 < DHID; k0 += 4) {
        // A fragment (16x4 f32): lanes 0-15 -> K=k0,k0+1 ; lanes 16-31 -> K=k0+2,k0+3
        v2f a;
        a.x = xr[k0 + 2 * hi + 0];
        a.y = xr[k0 + 2 * hi + 1];
        // B fragment (4x16 f32) per column tile: column = lrow, K striped as A
        const float* wr0 = W + (size_t)(k0 + 2 * hi + 0) * DHID;
        const float* wr1 = W + (size_t)(k0 + 2 * hi + 1) * DHID;
        v2f b0, b1, b2, b3;
        b0.x = wr0[lrow +  0]; b0.y = wr1[lrow +  0];
        b1.x = wr0[lrow + 16]; b1.y = wr1[lrow + 16];
        b2.x = wr0[lrow + 32]; b2.y = wr1[lrow + 32];
        b3.x = wr0[lrow + 48]; b3.y = wr1[lrow + 48];

        acc0 = __builtin_amdgcn_wmma_f32_16x16x4_f32(false, a, false, b0, (short)0, acc0, false, false);
        acc1 = __builtin_amdgcn_wmma_f32_16x16x4_f32(false, a, false, b1, (short)0, acc1, false, false);
        acc2 = __builtin_amdgcn_wmma_f32_16x16x4_f32(false, a, false, b2, (short)0, acc2, false, false);
        acc3 = __builtin_amdgcn_wmma_f32_16x16x4_f32(false, a, false, b3, (short)0, acc3, false, false);
    }

    // C/D layout: VGPR r -> row = row0 + r + 8*hi, col = lrow (+16 per tile)
#pragma unroll
    for (int r = 0; r < 8; ++r) {
        int row = row0 + r + 8 * hi;
        if (row < n) {
            float* hr = H + (size_t)row * DHID + lrow;
            hr[ 0] = acc0[r];
            hr[16] = acc1[r];
            hr[32] = acc2[r];
            hr[48] = acc3[r];
        }
    }
}

// ---------------------------------------------------------------------------
// Per-node attention scores: s_src[i] = h_i . a_src ; s_dst[i] = h_i . a_dst
// ---------------------------------------------------------------------------
__global__ __launch_bounds__(256)
void gat_scores(const float* __restrict__ H, const float* __restrict__ a_src,
                const float* __restrict__ a_dst,
                float* __restrict__ s_src, float* __restrict__ s_dst, int n)
{
    int i = blockIdx.x * blockDim.x + threadIdx.x;
    if (i >= n) return;
    const float4* hr = (const float4*)(H + (size_t)i * DHID);
    const float4* as = (const float4*)a_src;
    const float4* ad = (const float4*)a_dst;
    float ss = 0.f, sd = 0.f;
#pragma unroll
    for (int k = 0; k < DHID / 4; ++k) {
        float4 h4 = hr[k], a4 = as[k], d4 = ad[k];
        ss += h4.x * a4.x + h4.y * a4.y + h4.z * a4.z + h4.w * a4.w;
        sd += h4.x * d4.x + h4.y * d4.y + h4.z * d4.z + h4.w * d4.w;
    }
    s_src[i] = ss;
    s_dst[i] = sd;
}

// ---------------------------------------------------------------------------
// Per-layer init: zero accumulation matrix, m = -inf, denom = 0
// ---------------------------------------------------------------------------
__global__ __launch_bounds__(256)
void gat_init(float* __restrict__ O, float* __restrict__ m,
              float* __restrict__ denom, int n)
{
    int i = blockIdx.x * blockDim.x + threadIdx.x;
    if (i < n * DHID) O[i] = 0.f;
    if (i < n) { m[i] = -__builtin_huge_valf(); denom[i] = 0.f; }
}

__device__ __forceinline__ float leaky(float v) {
    return v > 0.f ? v : NEG_SLOPE * v;
}

__device__ __forceinline__ void atomicMaxFloat(float* addr, float v) {
    if (v >= 0.f) atomicMax((int*)addr, __float_as_int(v));
    else          atomicMin((unsigned int*)addr, (unsigned int)__float_as_int(v));
}

__device__ __forceinline__ void edge_endpoints(const int* __restrict__ srcs,
                                               const int* __restrict__ dsts,
                                               int e, int nE, int& s, int& d) {
    if (e < nE) { s = srcs[e]; d = dsts[e]; }
    else        { s = e - nE;  d = e - nE; }   // self loops appended at the end
}

// ---------------------------------------------------------------------------
// Pass 1: segment max of leaky_relu(s_src[src] + s_dst[dst]) into m[dst]
// ---------------------------------------------------------------------------
__global__ __launch_bounds__(256)
void gat_edge_max(const int* __restrict__ srcs, const int* __restrict__ dsts,
                  const float* __restrict__ s_src, const float* __restrict__ s_dst,
                  float* __restrict__ m, int nE, int n)
{
    int e = blockIdx.x * blockDim.x + threadIdx.x;
    if (e >= nE + n) return;
    int s, d; edge_endpoints(srcs, dsts, e, nE, s, d);
    atomicMaxFloat(&m[d], leaky(s_src[s] + s_dst[d]));
}

// ---------------------------------------------------------------------------
// Pass 2: denom[dst] += exp(e - m[dst])
// ---------------------------------------------------------------------------
__global__ __launch_bounds__(256)
void gat_edge_sum(const int* __restrict__ srcs, const int* __restrict__ dsts,
                  const float* __restrict__ s_src, const float* __restrict__ s_dst,
                  const float* __restrict__ m, float* __restrict__ denom,
                  int nE, int n)
{
    int e = blockIdx.x * blockDim.x + threadIdx.x;
    if (e >= nE + n) return;
    int s, d; edge_endpoints(srcs, dsts, e, nE, s, d);
    float v = leaky(s_src[s] + s_dst[d]);
    atomicAdd(&denom[d], __expf(v - m[d]));
}

// ---------------------------------------------------------------------------
// Pass 3: O[dst,:] += alpha * H[src,:]   (one wave per edge; 2 floats/lane)
// ---------------------------------------------------------------------------
__global__ __launch_bounds__(256)
void gat_edge_scatter(const int* __restrict__ srcs, const int* __restrict__ dsts,
                      const float* __restrict__ s_src, const float* __restrict__ s_dst,
                      const float* __restrict__ m, const float* __restrict__ denom,
                      const float* __restrict__ H, float* __restrict__ O,
                      int nE, int n)
{
    int t = blockIdx.x * blockDim.x + threadIdx.x;
    int e = t >> 5;
    int lane = t & 31;
    if (e >= nE + n) return;
    int s, d; edge_endpoints(srcs, dsts, e, nE, s, d);
    float v = leaky(s_src[s] + s_dst[d]);
    float w = __expf(v - m[d]) / denom[d];
    float2 h2 = ((const float2*)(H + (size_t)s * DHID))[lane];
    float* o = O + (size_t)d * DHID + lane * 2;
    atomicAdd(&o[0], w * h2.x);
    atomicAdd(&o[1], w * h2.y);
}

// ---------------------------------------------------------------------------
// Epilogue: O = (O + b), optional ReLU (in place)
// ---------------------------------------------------------------------------
__global__ __launch_bounds__(256)
void gat_bias_act(float* __restrict__ O, const float* __restrict__ b,
                  int n, int do_relu)
{
    int i = blockIdx.x * blockDim.x + threadIdx.x;
    if (i >= n * DHID) return;
    float v = O[i] + b[i & (DHID - 1)];
    if (do_relu) v = fmaxf(v, 0.f);
    O[i] = v;
}

// ---------------------------------------------------------------------------
// Mean pooling by graph id
// ---------------------------------------------------------------------------
__global__ __launch_bounds__(256)
void pool_init(float* __restrict__ pooled, float* __restrict__ counts)
{
    int i = blockIdx.x * blockDim.x + threadIdx.x;
    if (i < N_GRAPHS * DHID) pooled[i] = 0.f;
    if (i < N_GRAPHS) counts[i] = 0.f;
}

__global__ __launch_bounds__(256)
void pool_accum(const float* __restrict__ H, const int* __restrict__ batch,
                float* __restrict__ pooled, float* __restrict__ counts, int n)
{
    int i = blockIdx.x * blockDim.x + threadIdx.x;
    if (i >= n * DHID) return;
    int node = i >> 6, dcol = i & 63;
    int g = batch[node];
    atomicAdd(&pooled[g * DHID + dcol], H[i]);
    if (dcol == 0) atomicAdd(&counts[g], 1.f);
}

__global__ __launch_bounds__(256)
void pool_final(const float* __restrict__ pooled, const float* __restrict__ counts,
                float* __restrict__ out)
{
    int i = blockIdx.x * blockDim.x + threadIdx.x;
    if (i >= N_GRAPHS * DHID) return;
    out[i] = pooled[i] / fmaxf(counts[i >> 6], 1.f);
}

// ---------------------------------------------------------------------------
extern "C" void kernel_launch(void* const* d_in, const int* in_sizes, int n_in,
                              void* d_out, int out_size, void* d_ws, size_t ws_size,
                              hipStream_t stream)
{
    (void)in_sizes; (void)n_in; (void)out_size; (void)ws_size;

    const int N = N_NODES, E = N_EDGES;

    const float* x        = (const float*)d_in[0];
    const int*   adj      = (const int*)d_in[1];     // [2,E]: row0=src, row1=dst
    const int*   batch    = (const int*)d_in[2];
    const float* W_enc    = (const float*)d_in[3];
    const float* as_enc   = (const float*)d_in[4];
    const float* ad_enc   = (const float*)d_in[5];
    const float* b_enc    = (const float*)d_in[6];
    const float* W_proc   = (const float*)d_in[7];   // [2,D,D]
    const float* as_proc  = (const float*)d_in[8];   // [2,D]
    const float* ad_proc  = (const float*)d_in[9];
    const float* b_proc   = (const float*)d_in[10];

    const int* srcs = adj;
    const int* dsts = adj + E;

    // workspace carve
    float* ws     = (float*)d_ws;
    float* hmat   = ws;                                  // N*D
    float* omat   = hmat + (size_t)N * DHID;             // N*D (aliases layer input)
    float* ssrc   = omat + (size_t)N * DHID;             // N
    float* sdst   = ssrc + N;                            // N
    float* mbuf   = sdst + N;                            // N
    float* dbuf   = mbuf + N;                            // N
    float* pooled = dbuf + N;                            // G*D
    float* counts = pooled + N_GRAPHS * DHID;            // G

    // layer schedule: enc, proc0, proc1, proc0, proc1
    const float* lW[5]  = {W_enc,  W_proc,  W_proc + DHID * DHID, W_proc,  W_proc + DHID * DHID};
    const float* lAS[5] = {as_enc, as_proc, as_proc + DHID,       as_proc, as_proc + DHID};
    const float* lAD[5] = {ad_enc, ad_proc, ad_proc + DHID,       ad_proc, ad_proc + DHID};
    const float* lB[5]  = {b_enc,  b_proc,  b_proc + DHID,        b_proc,  b_proc + DHID};
    const int    lRelu[5] = {1, 1, 1, 1, 0};

    const int nEdgeTot  = E + N;
    const int gemmBlks  = ((N + 15) / 16 + 7) / 8;          // 8 waves/block, 16 rows/wave
    const int nodeBlks  = (N + 255) / 256;
    const int ndBlks    = (N * DHID + 255) / 256;
    const int edgeBlks  = (nEdgeTot + 255) / 256;
    const int scatBlks  = (int)(((long long)nEdgeTot * 32 + 255) / 256);

    const float* in = x;
    for (int l = 0; l < 5; ++l) {
        gat_gemm_wmma<<<gemmBlks, 256, 0, stream>>>(in, lW[l], hmat, N);
        gat_scores<<<nodeBlks, 256, 0, stream>>>(hmat, lAS[l], lAD[l], ssrc, sdst, N);
        gat_init<<<ndBlks, 256, 0, stream>>>(omat, mbuf, dbuf, N);
        gat_edge_max<<<edgeBlks, 256, 0, stream>>>(srcs, dsts, ssrc, sdst, mbuf, E, N);
        gat_edge_sum<<<edgeBlks, 256, 0, stream>>>(srcs, dsts, ssrc, sdst, mbuf, dbuf, E, N);
        gat_edge_scatter<<<scatBlks, 256, 0, stream>>>(srcs, dsts, ssrc, sdst, mbuf, dbuf,
                                                       hmat, omat, E, N);
        gat_bias_act<<<ndBlks, 256, 0, stream>>>(omat, lB[l], N, lRelu[l]);
        in = omat;   // epilogue result is next layer's input; GEMM frees it for reuse
    }

    pool_init<<<(N_GRAPHS * DHID + 255) / 256, 256, 0, stream>>>(pooled, counts);
    pool_accum<<<ndBlks, 256, 0, stream>>>(omat, batch, pooled, counts, N);
    pool_final<<<(N_GRAPHS * DHID + 255) / 256, 256, 0, stream>>>(pooled, counts, (float*)d_out);
}